// NetRGCN_20822001451274
// MI455X (gfx1250) — compile-verified
//
#include <hip/hip_runtime.h>
#include <hip/hip_bf16.h>
#include <math.h>

// ---------------------------------------------------------------------------
// RGCN head shortcut: output depends only on h[0]. See analysis.
// Workspace layout (floats): [0..256)  = x1
//                            [256]     = (int) match counter
//                            [257..257+CAP) = (int) matching edge ids
// ---------------------------------------------------------------------------

#define X1_OFF   0
#define LIST_OFF 256
#define CAP      2048
#define F_DIM    256

typedef __attribute__((ext_vector_type(2))) float v2f;
typedef __attribute__((ext_vector_type(8))) float v8f;

__global__ void k_init(float* ws) {
    ((int*)(ws + LIST_OFF))[0] = 0;
}

// Find all edges with dst == 0, compact their ids into a list.
__global__ void k_scan(const int* __restrict__ ei, float* ws, int nE) {
    int e = blockIdx.x * blockDim.x + threadIdx.x;
    if (e >= nE) return;
    if (ei[nE + e] == 0) {                 // dst row of edge_index
        int* wsi = (int*)(ws + LIST_OFF);
        int pos = atomicAdd(&wsi[0], 1);
        if (pos < CAP) wsi[1 + pos] = e;
    }
}

// Single block: sort the (tiny) match list for bitwise determinism, build
// per-relation sums & counts, then
//   x1 = relu( sum_b (sum_r comp[r,b]*sumx_r/cnt_r') @ basis_b
//              + x[0] @ root + bias )
__global__ void __launch_bounds__(256) k_x1(
        const int* __restrict__ ei, const int* __restrict__ et,
        const float* __restrict__ x, const float* __restrict__ comp,
        const float* __restrict__ basis, const float* __restrict__ root,
        const float* __restrict__ bias, float* ws, int nE)
{
    __shared__ int   sK;
    __shared__ int   sE[CAP];
    __shared__ float sSum[5][F_DIM];
    __shared__ float sU[5][F_DIM];
    __shared__ float sX0[F_DIM];
    __shared__ float sInv[5];

    const int tid = threadIdx.x;
    int* wsi = (int*)(ws + LIST_OFF);
    if (tid == 0) { int K = wsi[0]; sK = (K > CAP) ? CAP : K; }
    __syncthreads();
    const int K = sK;
    for (int i = tid; i < K; i += 256) sE[i] = wsi[1 + i];
    __syncthreads();
    if (tid == 0) {                        // insertion sort, K ~ 16
        for (int i = 1; i < K; ++i) {
            int v = sE[i]; int j = i - 1;
            while (j >= 0 && sE[j] > v) { sE[j + 1] = sE[j]; --j; }
            sE[j + 1] = v;
        }
    }
    __syncthreads();

    float acc[5] = {0.f, 0.f, 0.f, 0.f, 0.f};
    int   cnt[5] = {0, 0, 0, 0, 0};
    for (int i = 0; i < K; ++i) {
        int e = sE[i];
        int r = et[e];
        int s = ei[e];                     // src row of edge_index
        acc[r] += x[(size_t)s * F_DIM + tid];
        cnt[r] += 1;
    }
    for (int r = 0; r < 5; ++r) sSum[r][tid] = acc[r];
    if (tid < 5) sInv[tid] = 1.0f / fmaxf((float)cnt[tid], 1.0f);
    sX0[tid] = x[tid];                     // x[0, :]
    __syncthreads();

    for (int b = 0; b < 5; ++b) {
        float u = 0.f;
        for (int r = 0; r < 5; ++r)
            u += comp[r * 5 + b] * sInv[r] * sSum[r][tid];
        sU[b][tid] = u;
    }
    __syncthreads();

    float v = bias[tid];
    for (int b = 0; b < 5; ++b) {
        const float* B = basis + (size_t)b * F_DIM * F_DIM;
        for (int d = 0; d < F_DIM; ++d)
            v += sU[b][d] * B[(size_t)d * F_DIM + tid];
    }
    for (int d = 0; d < F_DIM; ++d)
        v += sX0[d] * root[(size_t)d * F_DIM + tid];

    ws[X1_OFF + tid] = fmaxf(v, 0.f);
}

// logits[row] = W[row,:] . x1 + b[row], via V_WMMA_F32_16X16X4_F32.
// One wave per 16-row tile; B column 0 holds the x1 chunk; D column 0
// (lanes 0 and 16) holds the 16 dot products.
__global__ void __launch_bounds__(256) k_gemv_wmma(
        const float* __restrict__ W, const float* __restrict__ bvec,
        const float* __restrict__ ws, float* __restrict__ out, int nrows)
{
    __shared__ float sX[F_DIM];
    const int tid = threadIdx.x;
    sX[tid] = ws[X1_OFF + tid];
    __syncthreads();

    const int wave = tid >> 5;
    const int lane = tid & 31;
    const int tile = blockIdx.x * 8 + wave;
    const int base = tile * 16;
    if (base >= nrows) return;             // wave-uniform: EXEC stays full

    const bool hi = lane >= 16;            // lanes 16-31 carry K+2,K+3
    const int  m  = lane & 15;
    int rowA = base + m;
    if (rowA > nrows - 1) rowA = nrows - 1;     // clamp ragged tail (25000%16)
    const float* wr = W + (size_t)rowA * F_DIM + (hi ? 2 : 0);

    v8f c = {};
    for (int k0 = 0; k0 < F_DIM; k0 += 4) {
        v2f a; a.x = wr[k0]; a.y = wr[k0 + 1];          // A: 16x4 f32 tile
        v2f b = {0.f, 0.f};                              // B: col 0 = x1 chunk
        if (lane == 0)  { b.x = sX[k0];     b.y = sX[k0 + 1]; }
        if (lane == 16) { b.x = sX[k0 + 2]; b.y = sX[k0 + 3]; }
        c = __builtin_amdgcn_wmma_f32_16x16x4_f32(
                /*neg_a=*/false, a, /*neg_b=*/false, b,
                /*c_mod=*/(short)0, c, /*reuse_a=*/false, /*reuse_b=*/false);
    }

    // D col 0: lane 0 has M=0..7 in c[0..7]; lane 16 has M=8..15.
    if (lane == 0 || lane == 16) {
        const int mb = hi ? 8 : 0;
        for (int j = 0; j < 8; ++j) {
            int row = base + mb + j;
            if (row < nrows) out[row] = c[j] + bvec[row];
        }
    }
}

// In-place log-softmax: block 0 -> [0, ng), block 1 -> [ng, ng+ns).
__global__ void __launch_bounds__(256) k_lsm(float* out, int ng, int ns) {
    __shared__ float red[256];
    const int tid   = threadIdx.x;
    const int start = (blockIdx.x == 0) ? 0 : ng;
    const int len   = (blockIdx.x == 0) ? ng : ns;
    float* p = out + start;

    float m = -INFINITY;
    for (int i = tid; i < len; i += 256) m = fmaxf(m, p[i]);
    red[tid] = m; __syncthreads();
    for (int s = 128; s > 0; s >>= 1) {
        if (tid < s) red[tid] = fmaxf(red[tid], red[tid + s]);
        __syncthreads();
    }
    m = red[0]; __syncthreads();

    float sum = 0.f;
    for (int i = tid; i < len; i += 256) sum += expf(p[i] - m);
    red[tid] = sum; __syncthreads();
    for (int s = 128; s > 0; s >>= 1) {
        if (tid < s) red[tid] += red[tid + s];
        __syncthreads();
    }
    const float lse = m + logf(red[0]);
    __syncthreads();

    for (int i = tid; i < len; i += 256) p[i] -= lse;
}

extern "C" void kernel_launch(void* const* d_in, const int* in_sizes, int n_in,
                              void* d_out, int out_size, void* d_ws, size_t ws_size,
                              hipStream_t stream) {
    const float* x     = (const float*)d_in[0];   // [50000,256]
    const int*   ei    = (const int*)  d_in[1];   // [2,800000]
    const int*   et    = (const int*)  d_in[2];   // [800000]
    const float* comp  = (const float*)d_in[3];   // [5,5]
    const float* basis = (const float*)d_in[4];   // [5,256,256]
    const float* root  = (const float*)d_in[5];   // [256,256]
    const float* bias  = (const float*)d_in[6];   // [256]
    const float* Wg    = (const float*)d_in[7];   // [20000,256]
    const float* bg    = (const float*)d_in[8];   // [20000]
    const float* Wsn   = (const float*)d_in[9];   // [25000,256]
    const float* bs    = (const float*)d_in[10];  // [25000]
    float* out = (float*)d_out;
    float* ws  = (float*)d_ws;

    const int nE = in_sizes[2];
    const int ng = in_sizes[8];
    const int ns = in_sizes[10];

    k_init<<<1, 1, 0, stream>>>(ws);
    k_scan<<<(nE + 255) / 256, 256, 0, stream>>>(ei, ws, nE);
    k_x1<<<1, 256, 0, stream>>>(ei, et, x, comp, basis, root, bias, ws, nE);

    const int tg = (ng + 15) / 16;   // tiles for global head
    const int ts = (ns + 15) / 16;   // tiles for sense head
    k_gemv_wmma<<<(tg + 7) / 8, 256, 0, stream>>>(Wg,  bg, ws, out,      ng);
    k_gemv_wmma<<<(ts + 7) / 8, 256, 0, stream>>>(Wsn, bs, ws, out + ng, ns);

    k_lsm<<<2, 256, 0, stream>>>(out, ng, ns);
}